// CSPNAccelerate_14929306321555
// MI455X (gfx1250) — compile-verified
//
#include <hip/hip_runtime.h>
#include <hip/hip_bf16.h>

// CSPN 3x3 accelerate: out[b,0,i,j] = sum_t kernel[b,t,i,j] * patch_t
// patch_t = input[b,0,i+dr-1,j+dc-1] (zero pad), center tap (t==4) = input0[b,0,i,j]
//
// Memory-bound (~0.35 flop/byte, ~236 MB compulsory traffic -> ~10us at 23.3 TB/s).
// Strategy: NT-hint the 177MB kernel stream + input0 + output (keep reused input in
// L2), stage the reused input tile into LDS via CDNA5 async global->LDS loads.

typedef float v4f __attribute__((ext_vector_type(4)));
typedef int   v4i __attribute__((ext_vector_type(4)));

typedef __attribute__((address_space(1))) v4i gv4i;  // global int4
typedef __attribute__((address_space(3))) v4i lv4i;  // LDS int4

#define BSZ      16
#define IMH      480
#define IMW      640
#define HW       (IMH * IMW)
#define ROWS     6                    // output rows per block
#define LROWS    (ROWS + 2)           // staged rows incl. halo
#define COL0     4                    // LDS column of image column 0 (16B alignment)
#define LDSW     (COL0 + IMW + 4)     // 648 floats per LDS row
#define NTHREADS 160                  // 5 waves; 160*4 = 640 pixels = full row

#if defined(__has_builtin)
#  if __has_builtin(__builtin_amdgcn_global_load_async_to_lds_b128)
#    define HAVE_ASYNC_BUILTIN 1
#  endif
#  if __has_builtin(__builtin_amdgcn_s_wait_asynccnt)
#    define HAVE_WAIT_BUILTIN 1
#  endif
#endif

// 16-byte async copy global -> LDS (CDNA5 GLOBAL_LOAD_ASYNC_TO_LDS_B128, ASYNCcnt)
__device__ __forceinline__ void async_copy16(const float* __restrict__ g, float* l) {
#if defined(HAVE_ASYNC_BUILTIN)
  // Param 1 (per clang diagnostic): int4* in the global (__device__/AS1) space.
  // Global AS1 pointer representation == flat 64-bit address.
  // LDS AS3 pointer representation == flat shared address truncated to 32 bits
  // (ISA 10.2: the LDS address is the low 32 bits).
  gv4i* gsrc = (gv4i*)(unsigned long long)g;
  lv4i* ldst = (lv4i*)(unsigned)(unsigned long long)l;
  __builtin_amdgcn_global_load_async_to_lds_b128(gsrc, ldst, 0, 0);
#else
  unsigned lds_addr = (unsigned)(unsigned long long)l;
  asm volatile("global_load_async_to_lds_b128 %0, %1, off"
               :: "v"(lds_addr), "v"(g)
               : "memory");
#endif
}

__device__ __forceinline__ void wait_async0() {
#if defined(HAVE_WAIT_BUILTIN)
  __builtin_amdgcn_s_wait_asynccnt(0);
#else
  asm volatile("s_wait_asynccnt 0" ::: "memory");
#endif
}

__global__ __launch_bounds__(NTHREADS) void cspn3x3_kernel(
    const float* __restrict__ kern,    // (bs, 9, H, W)
    const float* __restrict__ input,   // (bs, 1, H, W)
    const float* __restrict__ input0,  // (bs, 1, H, W)
    float* __restrict__ out)           // (bs, 1, H, W)
{
  __shared__ float tile[LROWS * LDSW];

  const int tid = threadIdx.x;
  const int b   = blockIdx.y;
  const int r0  = blockIdx.x * ROWS;

  const float* inp_b  = input  + (size_t)b * HW;
  const float* in0_b  = input0 + (size_t)b * HW;
  const float* kern_b = kern   + (size_t)b * 9 * HW;
  float*       out_b  = out    + (size_t)b * HW;

  // ---- stage input rows [r0-1, r0+ROWS] into LDS ----------------------------
  // 1) zero halo columns (image cols -1 and W) for every staged row.
  //    Disjoint from async-load targets, so no DS/async ordering hazard.
  if (tid < LROWS * 2) {
    const int lr = tid >> 1;
    const int c  = (tid & 1) ? (COL0 + IMW) : (COL0 - 1);
    tile[lr * LDSW + c] = 0.0f;
  }
  // 2) out-of-range rows: zero; valid rows: async B128 copy (uniform branch).
  for (int lr = 0; lr < LROWS; ++lr) {
    const int gr = r0 - 1 + lr;
    if (gr < 0 || gr >= IMH) {
      for (int c = tid; c < IMW; c += NTHREADS)
        tile[lr * LDSW + COL0 + c] = 0.0f;
    } else {
      async_copy16(inp_b + (size_t)gr * IMW + 4 * tid,
                   &tile[lr * LDSW + COL0 + 4 * tid]);
    }
  }
  wait_async0();
  __syncthreads();

  // ---- compute ROWS output rows; 4 pixels (one aligned float4) per thread ---
  const int j0 = 4 * tid;
#pragma unroll
  for (int r = 0; r < ROWS; ++r) {
    const int    gr     = r0 + r;
    const size_t rowoff = (size_t)gr * IMW + j0;

    // 3x6 neighborhood covering taps for the 4 pixels (image cols j0-1 .. j0+4)
    float p[3][6];
#pragma unroll
    for (int dr = 0; dr < 3; ++dr)
#pragma unroll
      for (int c = 0; c < 6; ++c)
        p[dr][c] = tile[(r + dr) * LDSW + (COL0 - 1) + j0 + c];

    const v4f x0 = __builtin_nontemporal_load((const v4f*)(in0_b + rowoff));

    float a0 = 0.0f, a1 = 0.0f, a2 = 0.0f, a3 = 0.0f;
#pragma unroll
    for (int t = 0; t < 9; ++t) {
      const v4f kv =
          __builtin_nontemporal_load((const v4f*)(kern_b + (size_t)t * HW + rowoff));
      if (t == 4) {  // center tap replaced by input0
        a0 += kv.x * x0.x;
        a1 += kv.y * x0.y;
        a2 += kv.z * x0.z;
        a3 += kv.w * x0.w;
      } else {
        const int dr = t / 3, dc = t % 3;
        a0 += kv.x * p[dr][dc + 0];
        a1 += kv.y * p[dr][dc + 1];
        a2 += kv.z * p[dr][dc + 2];
        a3 += kv.w * p[dr][dc + 3];
      }
    }
    v4f res;
    res.x = a0; res.y = a1; res.z = a2; res.w = a3;
    __builtin_nontemporal_store(res, (v4f*)(out_b + rowoff));
  }
}

extern "C" void kernel_launch(void* const* d_in, const int* in_sizes, int n_in,
                              void* d_out, int out_size, void* d_ws, size_t ws_size,
                              hipStream_t stream) {
  const float* kern   = (const float*)d_in[0];  // (bs, 9, H, W)
  const float* input  = (const float*)d_in[1];  // (bs, 1, H, W)
  const float* input0 = (const float*)d_in[2];  // (bs, 1, H, W)
  float*       out    = (float*)d_out;

  const int bs = in_sizes[1] / HW;              // 16 per reference setup
  dim3 grid(IMH / ROWS, bs);                    // (80, 16)
  cspn3x3_kernel<<<grid, NTHREADS, 0, stream>>>(kern, input, input0, out);
}